// CustomAttentionLayer_47785806135878
// MI455X (gfx1250) — compile-verified
//
#include <hip/hip_runtime.h>
#include <hip/hip_bf16.h>

#define B_DIM 32
#define T_DIM 4096
#define F_DIM 256
#define K_TOP 409          // max(1, int(T * 0.1))
#define EMPH  1.5f

typedef __attribute__((ext_vector_type(2))) float v2f;
typedef __attribute__((ext_vector_type(8))) float v8f;

// ---------------------------------------------------------------------------
// Pass 1: e[r] = tanh(dot(x[r,:], W) + bias) for r in [0, B*T)
// GEMV done as WMMA f32 16x16x4 with W replicated across the 16 N columns.
// One wave handles 16 rows; 8 waves per 256-thread block -> 128 rows/block.
// A layout (ISA 7.12.2, 32-bit A 16x4): lanes 0-15 hold K=0,1; lanes 16-31 K=2,3.
// ---------------------------------------------------------------------------
__global__ __launch_bounds__(256) void scores_wmma(const float* __restrict__ x,
                                                   const float* __restrict__ W,
                                                   const float* __restrict__ bias,
                                                   float* __restrict__ e)
{
    __shared__ float sW[F_DIM];
    sW[threadIdx.x] = W[threadIdx.x];
    __syncthreads();

    const int  lane = threadIdx.x & 31;
    const int  wave = threadIdx.x >> 5;
    const long t0   = ((long)blockIdx.x * 8 + wave) * 16;
    const int  m    = lane & 15;
    const int  hi   = lane >> 4;              // 0 -> K={0,1}, 1 -> K={2,3}
    const float* rowp = x + (t0 + m) * F_DIM;

    v8f acc = {};
    #pragma unroll 4
    for (int f = 0; f < F_DIM; f += 4) {
        v2f a, bb;
        a.x  = rowp[f + 2 * hi + 0];
        a.y  = rowp[f + 2 * hi + 1];
        bb.x = sW[f + 2 * hi + 0];            // B[k,n] = W[f+k] for every column n
        bb.y = sW[f + 2 * hi + 1];
        acc = __builtin_amdgcn_wmma_f32_16x16x4_f32(false, a, false, bb,
                                                    (short)0, acc, false, false);
    }

    const float bv = bias[0];
    // C/D layout: lanes 0-15 hold M=0..7 in VGPR 0..7, lanes 16-31 hold M=8..15.
    if (lane == 0) {
        #pragma unroll
        for (int i = 0; i < 8; ++i) e[t0 + i] = tanhf(acc[i] + bv);
    } else if (lane == 16) {
        #pragma unroll
        for (int i = 0; i < 8; ++i) e[t0 + 8 + i] = tanhf(acc[i] + bv);
    }
}

// ---------------------------------------------------------------------------
// Pass 2: per-batch softmax over T plus top-K_TOP emphasis.
// One block per batch. e[b,*] staged in LDS (16 KB). Threshold = K-th largest
// value, found by 32-step binary search on the monotone IEEE sort key.
// Overwrites e with the final weight w.
// ---------------------------------------------------------------------------
__device__ __forceinline__ unsigned sortkey(float f) {
    unsigned u = __float_as_uint(f);
    return (u & 0x80000000u) ? ~u : (u | 0x80000000u);
}

__global__ __launch_bounds__(256) void softmax_topk(float* __restrict__ ew)
{
    __shared__ float se[T_DIM];               // 16 KB
    __shared__ float red[256];
    __shared__ int   cnt;
    const int tid = threadIdx.x;
    float* ebat = ew + (long)blockIdx.x * T_DIM;

    float lmax = -3.402823466e+38f;
    #pragma unroll
    for (int i = 0; i < 16; ++i) {
        float v = ebat[tid + 256 * i];
        se[tid + 256 * i] = v;
        lmax = fmaxf(lmax, v);
    }
    red[tid] = lmax; __syncthreads();
    for (int s = 128; s > 0; s >>= 1) {
        if (tid < s) red[tid] = fmaxf(red[tid], red[tid + s]);
        __syncthreads();
    }
    const float gmax = red[0]; __syncthreads();

    float lsum = 0.f;
    #pragma unroll
    for (int i = 0; i < 16; ++i) lsum += expf(se[tid + 256 * i] - gmax);
    red[tid] = lsum; __syncthreads();
    for (int s = 128; s > 0; s >>= 1) {
        if (tid < s) red[tid] += red[tid + s];
        __syncthreads();
    }
    const float Z = red[0]; __syncthreads();

    // binary search for the K_TOP-th largest sort key (uniform across block)
    unsigned lo = 0u, hiv = 0xFFFFFFFFu;
    while (lo < hiv) {
        unsigned mid = lo + ((hiv - lo) >> 1) + 1u;
        if (tid == 0) cnt = 0;
        __syncthreads();
        int c = 0;
        #pragma unroll
        for (int i = 0; i < 16; ++i)
            c += (sortkey(se[tid + 256 * i]) >= mid) ? 1 : 0;
        atomicAdd(&cnt, c);
        __syncthreads();
        const int total = cnt;
        __syncthreads();                      // protect cnt before next reset
        if (total >= K_TOP) lo = mid; else hiv = mid - 1u;
    }
    const unsigned thresh = lo;

    const float inv = 1.0f / Z;
    #pragma unroll
    for (int i = 0; i < 16; ++i) {
        float v = se[tid + 256 * i];
        float a = expf(v - gmax) * inv;
        if (sortkey(v) >= thresh) a *= EMPH;
        ebat[tid + 256 * i] = a;
    }
}

// ---------------------------------------------------------------------------
// Pass 3: out[b,f] = sum_t w[b,t] * x[b,t,f].  Second streaming pass over x
// (mostly 192MB-L2 resident). Grid (B, T/256); partials combined with
// global f32 atomics. global_prefetch_b8 pulls the stream 32 rows (32 KB)
// ahead; speculative prefetch past the buffer end is dropped by hardware.
// ---------------------------------------------------------------------------
__global__ void zero_out(float* __restrict__ out)
{
    out[blockIdx.x * blockDim.x + threadIdx.x] = 0.0f;
}

__global__ __launch_bounds__(256) void weighted_sum(const float* __restrict__ x,
                                                    const float* __restrict__ w,
                                                    float* __restrict__ out)
{
    __shared__ float sw[256];
    const int  tid   = threadIdx.x;
    const long tbase = (long)blockIdx.x * T_DIM + (long)blockIdx.y * 256;
    sw[tid] = w[tbase + tid];
    __syncthreads();

    const float* xp = x + tbase * F_DIM + tid;
    float acc = 0.0f;
    #pragma unroll 8
    for (int t = 0; t < 256; ++t) {
        if ((t & 7) == 0)                      // once per unrolled group
            __builtin_prefetch(xp + (long)(t + 32) * F_DIM, 0, 0);
        acc += sw[t] * xp[(long)t * F_DIM];
    }

    atomicAdd(&out[blockIdx.x * F_DIM + tid], acc);
}

// ---------------------------------------------------------------------------
extern "C" void kernel_launch(void* const* d_in, const int* in_sizes, int n_in,
                              void* d_out, int out_size, void* d_ws, size_t ws_size,
                              hipStream_t stream)
{
    const float* x    = (const float*)d_in[0];   // [B, T, F]
    const float* W    = (const float*)d_in[1];   // [F, 1]
    const float* bias = (const float*)d_in[2];   // [1]
    float* out = (float*)d_out;                  // [B, 1, F]
    float* ws  = (float*)d_ws;                   // B*T floats (e -> w), 512 KB

    scores_wmma <<<dim3((B_DIM * T_DIM) / 128), 256, 0, stream>>>(x, W, bias, ws);
    softmax_topk<<<dim3(B_DIM),                 256, 0, stream>>>(ws);
    zero_out    <<<dim3((B_DIM * F_DIM) / 256), 256, 0, stream>>>(out);
    weighted_sum<<<dim3(B_DIM, T_DIM / 256),    256, 0, stream>>>(x, ws, out);
}